// Criterion_28278064676985
// MI455X (gfx1250) — compile-verified
//
#include <hip/hip_runtime.h>
#include <math.h>

// ---------------- problem constants (match reference) ----------------
#define N        4096
#define D        1024
#define POS_W    2.0f
#define NEG_W    40.0f
#define MARGIN   0.1f
#define THRESH   0.5f
#define BIG      1e9f

// ---------------- kernel geometry ----------------
#define NROWS          16                // anchor rows per workgroup (one WMMA M-tile)
#define NWAVES         8                 // waves per block (wave32)
#define BLOCK          (NWAVES * 32)     // 256 threads
#define TILES_PER_WAVE (N / 16 / NWAVES) // 32 column tiles of 16 per wave

typedef __attribute__((ext_vector_type(2))) float    v2f;
typedef __attribute__((ext_vector_type(8))) float    v8f;
typedef __attribute__((ext_vector_type(4))) unsigned u32x4;  // TDM D# group 0
typedef __attribute__((ext_vector_type(8))) unsigned u32x8;  // TDM D# group 1

// LDS byte offset of the staged-labels region (dynamic LDS starts at offset 0)
#define SLAB_LDS_OFF ((unsigned)((NROWS * N + 2 * BLOCK + 2 * NROWS) * 4))

__global__ void ms_init(float* gsum, float* gcnt) {
    gsum[0] = 0.0f;
    gcnt[0] = 0.0f;
}

// Fused: sim strip (f32 WMMA, kept entirely in LDS) + mining + masked exp sums.
__global__ __launch_bounds__(BLOCK) void ms_loss_main(
    const float* __restrict__ batch,   // [N, D] f32, L2-normalized rows
    const int*   __restrict__ labels,  // [N]
    float* __restrict__ gsum,          // scalar accumulators in d_ws
    float* __restrict__ gcnt)
{
    extern __shared__ float smem[];
    float* strip = smem;                      // [NROWS][N]  = 256 KB sim strip
    float* red1  = strip + NROWS * N;         // [BLOCK]
    float* red2  = red1 + BLOCK;              // [BLOCK]
    float* rowA  = red2 + BLOCK;              // [NROWS]  min_pos holder
    float* rowB  = rowA + NROWS;              // [NROWS]  max_neg holder
    int*   slab  = (int*)(rowB + NROWS);      // [N] labels, filled by TDM

    const int tid  = threadIdx.x;
    const int lane = tid & 31;
    const int wave = tid >> 5;
    const int half = lane >> 4;     // 0: lanes 0-15, 1: lanes 16-31
    const int l16  = lane & 15;
    const int rowBase = blockIdx.x * NROWS;

    // -------- TDM: DMA all 4096 labels into LDS, overlapped with the GEMM ----
    // One wave issues a single tensor_load_to_lds (TDM ignores EXEC; one wave
    // => one DMA). D# per CDNA5 ISA:
    //   group0: count=1 | lds_addr | global_addr[56:0] | type=2
    //   group1: data_size=4B, tensor_dim0=tile_dim0=4096, dim1=1, stride0=4096
    if (tid < 32) {
        unsigned long long ga = (unsigned long long)(const void*)labels;
        u32x4 g0;
        g0.x = 1u;                                                 // count = 1
        g0.y = SLAB_LDS_OFF;                                       // lds_addr
        g0.z = (unsigned)ga;                                       // addr[31:0]
        g0.w = (unsigned)((ga >> 32) & 0x01FFFFFFu) | (2u << 30);  // addr[56:32]|type=2
        u32x8 g1;
        g1.s0 = 2u << 16;                         // data_size = 4 bytes
        g1.s1 = ((unsigned)N & 0xFFFFu) << 16;    // tensor_dim0[15:0]  (bits 63:48)
        g1.s2 = ((unsigned)N >> 16) | (1u << 16); // tensor_dim0[31:16] | tensor_dim1=1
        g1.s3 = (unsigned)N << 16;                // tile_dim0 = 4096   (bits 127:112)
        g1.s4 = 1u;                               // tile_dim1 = 1
        g1.s5 = (unsigned)N;                      // tensor_dim0_stride (bits 191:160)
        g1.s6 = ((unsigned)N & 0xFFFFu) << 16;    // tensor_dim1_stride[15:0]
        g1.s7 = 0u;
        asm volatile("tensor_load_to_lds %0, %1" :: "s"(g0), "s"(g1) : "memory");
    }

    // -------- Phase 1: sim strip via V_WMMA_F32_16X16X4_F32 --------
    // A fragment (16x4, f32): lane m=l16 holds K = 2*half + {0,1}  -> float2 load
    // B fragment (4x16, f32): lane n=l16 holds K = 2*half + {0,1} of batch row j
    // 4 N-tiles per K-pass: one A fragment feeds 4 independent wmma acc chains.
    const float* arow = batch + (size_t)(rowBase + l16) * D + half * 2;
    for (int t = 0; t < TILES_PER_WAVE; t += 4) {
        const int jBase = wave * (TILES_PER_WAVE * 16) + t * 16;
        const float* brow0 = batch + (size_t)(jBase +  0 + l16) * D + half * 2;
        const float* brow1 = batch + (size_t)(jBase + 16 + l16) * D + half * 2;
        const float* brow2 = batch + (size_t)(jBase + 32 + l16) * D + half * 2;
        const float* brow3 = batch + (size_t)(jBase + 48 + l16) * D + half * 2;
        if (t + 4 < TILES_PER_WAVE) {
            // global_prefetch_b8 of the next tile group's B rows
            __builtin_prefetch(batch + (size_t)(jBase + 64 + l16) * D, 0, 3);
        }
        v8f acc0 = {}, acc1 = {}, acc2 = {}, acc3 = {};
        #pragma unroll 4
        for (int kk = 0; kk < D; kk += 4) {
            v2f a  = *(const v2f*)(arow  + kk);
            v2f b0 = *(const v2f*)(brow0 + kk);
            v2f b1 = *(const v2f*)(brow1 + kk);
            v2f b2 = *(const v2f*)(brow2 + kk);
            v2f b3 = *(const v2f*)(brow3 + kk);
            // 8 args: (neg_a, A, neg_b, B, c_mod, C, reuse_a, reuse_b)
            acc0 = __builtin_amdgcn_wmma_f32_16x16x4_f32(false, a, false, b0, (short)0, acc0, false, false);
            acc1 = __builtin_amdgcn_wmma_f32_16x16x4_f32(false, a, false, b1, (short)0, acc1, false, false);
            acc2 = __builtin_amdgcn_wmma_f32_16x16x4_f32(false, a, false, b2, (short)0, acc2, false, false);
            acc3 = __builtin_amdgcn_wmma_f32_16x16x4_f32(false, a, false, b3, (short)0, acc3, false, false);
        }
        // C/D layout: VGPR v -> row (v + 8*half), col = jBase + l16
        const int r0 = half * 8;
        #pragma unroll
        for (int v = 0; v < 8; ++v) {
            strip[(r0 + v) * N + jBase +  0 + l16] = acc0[v];
            strip[(r0 + v) * N + jBase + 16 + l16] = acc1[v];
            strip[(r0 + v) * N + jBase + 32 + l16] = acc2[v];
            strip[(r0 + v) * N + jBase + 48 + l16] = acc3[v];
        }
    }
    // The issuing wave must retire its TDM before anyone reads slab[]
    if (tid < 32) __builtin_amdgcn_s_wait_tensorcnt(0);
    __syncthreads();

    // -------- Phase 2a: per-row hardest positive / negative --------
    const int r   = tid >> 4;   // 0..15 : row within strip
    const int sub = tid & 15;   // 16 threads cooperate per row
    const int myLab = slab[rowBase + r];

    float mn = BIG, mx = -BIG;
    for (int c = sub; c < N; c += 16) {
        float s   = strip[r * N + c];
        bool same = (slab[c] == myLab);
        bool self = (c == rowBase + r);
        if (same && !self) mn = fminf(mn, s);
        if (!same)         mx = fmaxf(mx, s);
    }
    red1[tid] = mn; red2[tid] = mx;
    __syncthreads();
    if (tid < NROWS) {
        float a = BIG, b = -BIG;
        #pragma unroll
        for (int i = 0; i < 16; ++i) {
            a = fminf(a, red1[tid * 16 + i]);
            b = fmaxf(b, red2[tid * 16 + i]);
        }
        rowA[tid] = a; rowB[tid] = b;
    }
    __syncthreads();

    // -------- Phase 2b: masked exp sums with mining thresholds --------
    const float minPos = rowA[r];
    const float maxNeg = rowB[r];
    float ps = 0.0f, ns = 0.0f;
    for (int c = sub; c < N; c += 16) {
        float s   = strip[r * N + c];
        bool same = (slab[c] == myLab);
        bool self = (c == rowBase + r);
        if (!same && (s + MARGIN > minPos)) ns += __expf(NEG_W * (s - THRESH));
        if (same && !self && (s - MARGIN < maxNeg)) ps += __expf(-POS_W * (s - THRESH));
    }
    __syncthreads();              // red1/red2 reuse
    red1[tid] = ps; red2[tid] = ns;
    __syncthreads();
    if (tid < NROWS) {
        float P = 0.0f, Ng = 0.0f;
        #pragma unroll
        for (int i = 0; i < 16; ++i) {
            P  += red1[tid * 16 + i];
            Ng += red2[tid * 16 + i];
        }
        // exp terms are strictly positive and never underflow to 0 here
        // (exponents bounded by 3 and 60), so sum>0 <=> any kept.
        if (P > 0.0f && Ng > 0.0f) {
            float per = log1pf(P) / POS_W + log1pf(Ng) / NEG_W;
            atomicAdd(gsum, per);
            atomicAdd(gcnt, 1.0f);
        }
    }
}

__global__ void ms_fin(const float* gsum, const float* gcnt, float* out) {
    out[0] = gsum[0] / fmaxf(gcnt[0], 1.0f);
}

extern "C" void kernel_launch(void* const* d_in, const int* in_sizes, int n_in,
                              void* d_out, int out_size, void* d_ws, size_t ws_size,
                              hipStream_t stream) {
    const float* batch  = (const float*)d_in[0];
    const int*   labels = (const int*)d_in[1];
    float* gsum = (float*)d_ws;        // re-zeroed every launch -> deterministic
    float* gcnt = gsum + 1;

    const size_t smem_bytes =
        (size_t)(NROWS * N + 2 * BLOCK + 2 * NROWS) * sizeof(float)  // strip + reductions
        + (size_t)N * sizeof(int);                                   // labels stage
    (void)hipFuncSetAttribute((const void*)ms_loss_main,
                              hipFuncAttributeMaxDynamicSharedMemorySize,
                              (int)smem_bytes);

    ms_init<<<1, 1, 0, stream>>>(gsum, gcnt);
    ms_loss_main<<<N / NROWS, BLOCK, smem_bytes, stream>>>(batch, labels, gsum, gcnt);
    ms_fin<<<1, 1, 0, stream>>>(gsum, gcnt, (float*)d_out);
}